// Hyper_Attention_71940702208269
// MI455X (gfx1250) — compile-verified
//
#include <hip/hip_runtime.h>

// Problem constants (from reference): B=2, N=4096, C=192, H=8, D=24
#define BATCH 2
#define SEQ   4096
#define CH    192
#define HEADS 8
#define HDIM  24
#define ROWS  (BATCH * SEQ)      // 8192 token rows

typedef __attribute__((ext_vector_type(2))) float v2f;
typedef __attribute__((ext_vector_type(8))) float v8f;

// ---------------------------------------------------------------------------
// Kernel A: wksum[h][c] = sum_d W_qkv[(C + h*24 + d)][c]   (k-block row sums)
// ---------------------------------------------------------------------------
__global__ __launch_bounds__(256) void wksum_kernel(const float* __restrict__ Wqkv,
                                                    float* __restrict__ wks) {
    int t = blockIdx.x * blockDim.x + threadIdx.x;   // HEADS*CH = 1536
    if (t >= HEADS * CH) return;
    int h = t / CH, c = t % CH;
    float s = 0.f;
#pragma unroll
    for (int d = 0; d < HDIM; ++d)
        s += Wqkv[(CH + h * HDIM + d) * CH + c];
    wks[h * CH + c] = s;
}

// ---------------------------------------------------------------------------
// Kernel B: ks[bn][h] = x[bn,:] . wksum[h,:]   (softmax logits, pre-scale)
// ---------------------------------------------------------------------------
__global__ __launch_bounds__(256) void ks_kernel(const float* __restrict__ x,
                                                 const float* __restrict__ wks,
                                                 float* __restrict__ ks) {
    int t = blockIdx.x * blockDim.x + threadIdx.x;   // ROWS*HEADS = 65536
    if (t >= ROWS * HEADS) return;
    int bn = t >> 3, h = t & 7;
    const float* xr = x + bn * CH;
    const float* wr = wks + h * CH;
    float s = 0.f;
#pragma unroll 4
    for (int c = 0; c < CH; ++c) s = fmaf(xr[c], wr[c], s);
    ks[t] = s;
}

// ---------------------------------------------------------------------------
// Kernel C: v = x @ Wv^T using V_WMMA_F32_16X16X4_F32.
// One wave (32 lanes) computes one 16x16 output tile; K=192 -> 48 WMMAs.
//   A frag (16x4 f32): lane l, vgpr j -> A[l%16][kbase + 2*(l/16) + j]
//   B frag (4x16 f32): lane l, vgpr j -> B[kbase + 2*(l/16) + j][l%16]
//     with B[k][n] = Wv[n][k]  (Wv rows are W_qkv rows 2C..3C-1)
//   D (16x16 f32, 8 vgprs): vgpr j, lanes 0-15 -> M=j, lanes 16-31 -> M=j+8
// ---------------------------------------------------------------------------
__global__ __launch_bounds__(32) void vgemm_wmma_kernel(const float* __restrict__ x,
                                                        const float* __restrict__ Wv,
                                                        float* __restrict__ v) {
    const int mt   = blockIdx.x;          // 0..511  (ROWS/16)
    const int nt   = blockIdx.y;          // 0..11   (CH/16)
    const int lane = threadIdx.x;         // 0..31
    const int mr   = lane & 15;
    const int kg   = (lane >> 4) << 1;    // 0 or 2

    const float* arow = x  + (mt * 16 + mr) * CH;   // A row m = mt*16 + mr
    const float* brow = Wv + (nt * 16 + mr) * CH;   // Wv row n = nt*16 + mr

    v8f acc = {};
#pragma unroll 4
    for (int k = 0; k < CH; k += 4) {
        v2f a = *(const v2f*)(arow + k + kg);
        v2f b = *(const v2f*)(brow + k + kg);
        acc = __builtin_amdgcn_wmma_f32_16x16x4_f32(
            /*neg_a=*/false, a, /*neg_b=*/false, b,
            /*c_mod=*/(short)0, acc, /*reuse_a=*/false, /*reuse_b=*/false);
    }

    const int mbase = mt * 16 + ((lane >> 4) << 3);
    const int ncol  = nt * 16 + mr;
#pragma unroll
    for (int j = 0; j < 8; ++j)
        v[(mbase + j) * CH + ncol] = acc[j];
}

// ---------------------------------------------------------------------------
// Kernel D: per (b,h): w = softmax_n(ks*scale);  o[b,h,d] = sum_n w_n v[b,n,h*24+d]
// One 256-thread block per (b,h). Two passes: max, then exp-sum + weighted acc.
// ---------------------------------------------------------------------------
__global__ __launch_bounds__(256) void attn_kernel(const float* __restrict__ ks,
                                                   const float* __restrict__ v,
                                                   const float* __restrict__ scale_p,
                                                   float* __restrict__ o) {
    const int bh  = blockIdx.x;           // 0..15
    const int b   = bh >> 3, h = bh & 7;
    const int tid = threadIdx.x;
    const float scale = scale_p[0];

    __shared__ float red[256];
    __shared__ float buf[25][260];        // 25 partial sums x 256 lanes (padded)

    // ---- pass 1: row max of logits ----
    float m = -3.402823466e38f;
    for (int n = tid; n < SEQ; n += 256)
        m = fmaxf(m, ks[(b * SEQ + n) * HEADS + h] * scale);
    red[tid] = m;
    __syncthreads();
    for (int s2 = 128; s2 > 0; s2 >>= 1) {
        if (tid < s2) red[tid] = fmaxf(red[tid], red[tid + s2]);
        __syncthreads();
    }
    m = red[0];
    __syncthreads();

    // ---- pass 2: exp-sum and weighted v accumulation ----
    float ssum = 0.f;
    float acc[HDIM];
#pragma unroll
    for (int d = 0; d < HDIM; ++d) acc[d] = 0.f;

    for (int n = tid; n < SEQ; n += 256) {
        float e = __expf(ks[(b * SEQ + n) * HEADS + h] * scale - m);
        ssum += e;
        const float* vr = v + (b * SEQ + n) * CH + h * HDIM;
#pragma unroll
        for (int d = 0; d < HDIM; ++d) acc[d] = fmaf(e, vr[d], acc[d]);
    }

    buf[0][tid] = ssum;
#pragma unroll
    for (int d = 0; d < HDIM; ++d) buf[d + 1][tid] = acc[d];
    __syncthreads();
    for (int s2 = 128; s2 > 0; s2 >>= 1) {
        if (tid < s2) {
#pragma unroll
            for (int j = 0; j < 25; ++j) buf[j][tid] += buf[j][tid + s2];
        }
        __syncthreads();
    }
    if (tid < HDIM)
        o[(b * HEADS + h) * HDIM + tid] = buf[tid + 1][0] / buf[0][0];
}

// ---------------------------------------------------------------------------
// Kernel E: y[b][j] = o[b,:] . W_proj[j,:] + b_proj[j]     (2 x 192 outputs)
// ---------------------------------------------------------------------------
__global__ __launch_bounds__(192) void proj_kernel(const float* __restrict__ o,
                                                   const float* __restrict__ Wp,
                                                   const float* __restrict__ bp,
                                                   float* __restrict__ y) {
    int b = blockIdx.x;                   // 0..1
    int j = threadIdx.x;                  // 0..191
    const float* orow = o + b * CH;
    const float* wr   = Wp + j * CH;
    float s = bp[j];
#pragma unroll 4
    for (int c = 0; c < CH; ++c) s = fmaf(orow[c], wr[c], s);
    y[b * CH + j] = s;
}

// ---------------------------------------------------------------------------
// Kernel F: broadcast y[b,:] to out[b,n,:] for all n (float4 stream write)
// ---------------------------------------------------------------------------
__global__ __launch_bounds__(256) void bcast_kernel(const float* __restrict__ y,
                                                    float4* __restrict__ out) {
    long i = (long)blockIdx.x * blockDim.x + threadIdx.x;   // ROWS*48 float4s
    if (i >= (long)ROWS * (CH / 4)) return;
    int row = (int)(i / (CH / 4));
    int b   = row >> 12;                  // row / SEQ
    int c4  = (int)(i % (CH / 4));
    const float4* y4 = (const float4*)y;
    out[i] = y4[b * (CH / 4) + c4];
}

// ---------------------------------------------------------------------------
extern "C" void kernel_launch(void* const* d_in, const int* in_sizes, int n_in,
                              void* d_out, int out_size, void* d_ws, size_t ws_size,
                              hipStream_t stream) {
    (void)in_sizes; (void)n_in; (void)out_size; (void)ws_size;

    const float* x     = (const float*)d_in[0];   // [B,N,C]
    const float* Wqkv  = (const float*)d_in[1];   // [3C,C]
    const float* Wproj = (const float*)d_in[2];   // [C,C]
    const float* bproj = (const float*)d_in[3];   // [C]
    const float* scale = (const float*)d_in[4];   // [1]
    float* out = (float*)d_out;                   // [B,N,C]

    // workspace layout
    char* ws = (char*)d_ws;
    float* v_buf  = (float*)(ws);                                   // ROWS*CH
    float* ks_buf = (float*)(ws + (size_t)ROWS * CH * 4);           // ROWS*HEADS
    float* wk_buf = (float*)(ws + (size_t)ROWS * CH * 4
                                + (size_t)ROWS * HEADS * 4);        // HEADS*CH
    float* o_buf  = wk_buf + HEADS * CH;                            // BATCH*CH
    float* y_buf  = o_buf + BATCH * CH;                             // BATCH*CH

    // A: k-block column sums
    wksum_kernel<<<(HEADS * CH + 255) / 256, 256, 0, stream>>>(Wqkv, wk_buf);
    // B: logits ks = x @ wksum^T
    ks_kernel<<<(ROWS * HEADS) / 256, 256, 0, stream>>>(x, wk_buf, ks_buf);
    // C: v = x @ Wv^T via f32 WMMA (Wv = rows 2C..3C-1 of W_qkv)
    dim3 gv(ROWS / 16, CH / 16);
    vgemm_wmma_kernel<<<gv, 32, 0, stream>>>(x, Wqkv + 2 * CH * CH, v_buf);
    // D: softmax over n + weighted reduction of v -> o[B,H,D]
    attn_kernel<<<BATCH * HEADS, 256, 0, stream>>>(ks_buf, v_buf, scale, o_buf);
    // E: output projection on the collapsed [B,C] vector
    proj_kernel<<<BATCH, CH, 0, stream>>>(o_buf, Wproj, bproj, y_buf);
    // F: broadcast to [B,N,C]
    bcast_kernel<<<(ROWS * (CH / 4)) / 256, 256, 0, stream>>>(y_buf, (float4*)out);
}